// ScaledDotProductAttention_1614907703360
// MI455X (gfx1250) — compile-verified
//
#include <hip/hip_runtime.h>
#include <math.h>

// ---------------------------------------------------------------------------
// Sizes: b=4, h=8, l=1024, d=64, D=h*d=512, BH=32, 4 scales k in {1,3,6,9}
// out = context (4*8*1024*64 = 2097152 f32)  ++  attn_out (4*8*4*1024 = 131072 f32)
//
// Pipeline (all f32, all GEMMs via v_wmma_f32_16x16x4_f32):
//  1) build DFT matrices C_D,S_D (512^2), C_L, -S_L (1024^2)
//  2) Tc = X*C_D, Ts = X*S_D          (X = Q/K viewed as (1024,512) per batch)
//  3) F  = C_L*Tc + (-S_L)*Ts         (= Re(FFT2), "s layout")
//  4) multi-scale windowed gather -> Qp, Kp  [bh][p][1024][64]
//  5) context = colsum(V) broadcast (softmax over p sums to 1 => Q,K drop out)
//  6) attn_out[bh][p][k] = sum_q softmax_p( SCALE * Qp_p[q,:].Kp_p[k,:] )
// ---------------------------------------------------------------------------

typedef float v2f __attribute__((ext_vector_type(2)));
typedef float v8f __attribute__((ext_vector_type(8)));

static __device__ __forceinline__ v8f v8f_zero() {
  v8f z = {0.f, 0.f, 0.f, 0.f, 0.f, 0.f, 0.f, 0.f};
  return z;
}

static __device__ __forceinline__ v8f wmma4(v2f a, v2f b, v8f c) {
  // D = A(16x4) * B(4x16) + C(16x16), fp32, wave32
  return __builtin_amdgcn_wmma_f32_16x16x4_f32(false, a, false, b, (short)0, c,
                                               false, false);
}

// ---------------------------------------------------------------------------
// 1) DFT matrix builder: C_N[n,j]=cos(2*pi*n*j/N).
//    S_L is stored NEGATED so pass-2 is a pure 2-product accumulation
//    (f32 WMMA NEG bits are C-only per the ISA, so we bake the sign in).
// ---------------------------------------------------------------------------
__global__ void build_dft_kernel(float* __restrict__ CD, float* __restrict__ SD,
                                 float* __restrict__ CL, float* __restrict__ SLn) {
  const int idx = blockIdx.x * blockDim.x + threadIdx.x;  // 0 .. 1048575
  const int n = idx >> 10, j = idx & 1023;
  const float TWO_PI = 6.28318530717958647692f;
  const int r = (n * j) & 1023;  // (n*j) mod 1024 (exact, power of two)
  const float ang = TWO_PI * ((float)r * (1.0f / 1024.0f));
  CL[idx] = cosf(ang);
  SLn[idx] = -sinf(ang);
  if (n < 512 && j < 512) {
    const int r2 = (n * j) & 511;
    const float a2 = TWO_PI * ((float)r2 * (1.0f / 512.0f));
    CD[n * 512 + j] = cosf(a2);
    SD[n * 512 + j] = sinf(a2);
  }
}

// ---------------------------------------------------------------------------
// 2,3) Batched GEMM:  C[z] = A1[z]*B1[z] (+ A2[z]*B2[z] if nprod==2)
//      Row-major, lda=Kd, ldb=ldc=N. Block tile 64x64 (4 waves x 16 rows),
//      K-step 32, LDS-staged with WMMA-fragment-friendly layouts:
//        As[64][36]      : A pair (kk,kk+1) contiguous -> one ds_load_b64
//        Bs[16][80][2]   : B[kk][n] at Bs[kk>>1][n][kk&1]; the (kk,kk+1)
//                          pair at column n is contiguous -> one ds_load_b64
// ---------------------------------------------------------------------------
__global__ __launch_bounds__(128) void gemm2_kernel(
    const float* __restrict__ A1, int a1s, const float* __restrict__ B1, int b1s,
    const float* __restrict__ A2, int a2s, const float* __restrict__ B2, int b2s,
    float* __restrict__ C, int cs, int M, int N, int Kd, int nprod) {
  __shared__ __align__(16) float As[64][36];
  __shared__ __align__(16) float Bs[16][80][2];
  const int tid = threadIdx.x;
  const int wave = tid >> 5, lane = tid & 31;
  const int z = blockIdx.z;
  const int m0 = blockIdx.y << 6, n0 = blockIdx.x << 6;
  const int mrow = (wave << 4) + (lane & 15);
  const int hoff = (lane >> 4) << 1;  // 0 or 2 : K-pair selector per half-wave

  v8f acc[4];
#pragma unroll
  for (int i = 0; i < 4; ++i) acc[i] = v8f_zero();

  for (int prod = 0; prod < nprod; ++prod) {
    const float* pA = prod ? (A2 + (size_t)z * a2s) : (A1 + (size_t)z * a1s);
    const float* pB = prod ? (B2 + (size_t)z * b2s) : (B1 + (size_t)z * b1s);
    for (int k0 = 0; k0 < Kd; k0 += 32) {
      // stage A 64x32 (512 float4 chunks) and B 32x64
#pragma unroll
      for (int c = tid; c < 512; c += 128) {
        const int r = c >> 3, kk = (c & 7) << 2;
        const float4 va = *(const float4*)(pA + (size_t)(m0 + r) * Kd + k0 + kk);
        As[r][kk] = va.x; As[r][kk + 1] = va.y; As[r][kk + 2] = va.z; As[r][kk + 3] = va.w;
      }
#pragma unroll
      for (int c = tid; c < 512; c += 128) {
        const int r = c >> 4, nn = (c & 15) << 2;
        const float4 vb = *(const float4*)(pB + (size_t)(k0 + r) * N + n0 + nn);
        const int rp = r >> 1, rl = r & 1;
        Bs[rp][nn][rl] = vb.x; Bs[rp][nn + 1][rl] = vb.y;
        Bs[rp][nn + 2][rl] = vb.z; Bs[rp][nn + 3][rl] = vb.w;
      }
      // prefetch next k-tile while this one is consumed (global_prefetch_b8)
      if (k0 + 32 < Kd) {
        __builtin_prefetch(pA + (size_t)(m0 + (tid & 63)) * Kd + (k0 + 32) +
                               ((tid >> 6) << 4), 0, 3);
        __builtin_prefetch(pB + (size_t)(k0 + 32 + (tid & 31)) * N + n0 +
                               ((tid >> 5) << 4), 0, 3);
      }
      __syncthreads();
#pragma unroll
      for (int ks = 0; ks < 8; ++ks) {
        const int kk = (ks << 2) + hoff;          // always even
        const v2f a = *(const v2f*)&As[mrow][kk]; // ds_load_b64
#pragma unroll
        for (int nt = 0; nt < 4; ++nt) {
          const v2f b = *(const v2f*)&Bs[kk >> 1][(nt << 4) + (lane & 15)][0];
          acc[nt] = wmma4(a, b, acc[nt]);
        }
      }
      __syncthreads();
    }
  }

  float* pC = C + (size_t)z * cs;
#pragma unroll
  for (int nt = 0; nt < 4; ++nt) {
    const int col = n0 + (nt << 4) + (lane & 15);
#pragma unroll
    for (int i = 0; i < 8; ++i) {
      const int row = m0 + (wave << 4) + i + ((lane >> 4) << 3);
      pC[(size_t)row * N + col] = acc[nt][i];
    }
  }
}

// ---------------------------------------------------------------------------
// 4) multi-scale: replicate the reference's pad/window/reshape scramble.
//    F: 4 batches of (1024,512) ("s layout"), P: [bh][p][1024][64]
// ---------------------------------------------------------------------------
__global__ __launch_bounds__(256) void multiscale_kernel(
    const float* __restrict__ F, const float* __restrict__ w1,
    const float* __restrict__ w3, const float* __restrict__ w6,
    const float* __restrict__ w9, const float* __restrict__ b1,
    const float* __restrict__ b3, const float* __restrict__ b6,
    const float* __restrict__ b9, float* __restrict__ P) {
  const int t = blockIdx.x * 256 + threadIdx.x;  // 2^23 threads total
  const int dd = t & 63;
  const int q = (t >> 6) & 1023;
  const int p = (t >> 16) & 3;
  const int h = (t >> 18) & 7;
  const int bb = t >> 21;
  const int kv[4] = {1, 3, 6, 9};
  const int k = kv[p];
  const float* wp = (p == 0) ? w1 : (p == 1) ? w3 : (p == 2) ? w6 : w9;
  const float* bp = (p == 0) ? b1 : (p == 1) ? b3 : (p == 2) ? b6 : b9;
  float acc = bp[0];
  const int Dk = 512 * k;
  const int lkd = 65536 * k;
  const float* Fb = F + (size_t)bb * 524288;
  for (int j2 = 0; j2 < k; ++j2) {
    const int flat = h * lkd + q * (k << 6) + (j2 << 6) + dd;
    const int i = flat / Dk;
    const int rem = flat - i * Dk;
    const int m = rem / k;
    const int jj = rem - m * k;
    const int ii = i + jj - k;
    const int mm = m - k;
    if ((unsigned)ii < 1024u && mm >= 0)
      acc += wp[j2] * Fb[(size_t)ii * 512 + mm];
  }
  P[(size_t)((((bb << 3) + h) << 2) + p) * 65536 + (q << 6) + dd] = acc;
}

// ---------------------------------------------------------------------------
// 5) context[b,h,q,:] = colsum(V[b,h])  (softmax over p sums to 1)
// ---------------------------------------------------------------------------
__global__ __launch_bounds__(256) void context_kernel(const float* __restrict__ V,
                                                      float* __restrict__ out) {
  __shared__ float part[4][64];
  __shared__ float colsum[64];
  const int bh = blockIdx.x;
  const int t = threadIdx.x;
  const int d = t & 63, g = t >> 6;
  const float* Vb = V + (size_t)bh * 65536;
  float s = 0.f;
  for (int r = g; r < 1024; r += 4) s += Vb[(r << 6) + d];
  part[g][d] = s;
  __syncthreads();
  if (t < 64) colsum[t] = part[0][t] + part[1][t] + part[2][t] + part[3][t];
  __syncthreads();
  const float cv = colsum[d];
  float* ob = out + (size_t)bh * 65536;
  for (int r = g; r < 1024; r += 4) ob[(r << 6) + d] = cv;
}

// ---------------------------------------------------------------------------
// 6) fused scores + softmax(over p) + sum-over-q.
//    Block = 128 thr (4 waves), wave w <-> scale p=w, k-tile of 16 per block.
//    Loop q-tiles of 16: each wave does one 16x16 wmma tile (K=64, 16 steps),
//    scores exchanged through LDS, per-element softmax over the 4 scales,
//    per-thread partial sums over q, final LDS reduction -> attn_out.
// ---------------------------------------------------------------------------
__global__ __launch_bounds__(128) void attn_kernel(const float* __restrict__ Qp,
                                                   const float* __restrict__ Kp,
                                                   float* __restrict__ attn_out) {
  __shared__ __align__(16) float KsS[4][16][68];  // [p][kc][d], padded
  __shared__ __align__(16) float QsS[4][16][68];  // [p][qr][d], padded
  __shared__ float Sx[4][16][17];   // scores exchange [p][qr][kc]
  __shared__ float red[4][16][8];   // final reduction
  const int tid = threadIdx.x;
  const int wave = tid >> 5, lane = tid & 31;
  const int kb = blockIdx.x, bh = blockIdx.y;
  const int k0 = kb << 4;
  const int n = lane & 15;
  const int hoff = (lane >> 4) << 1;
  const int kc = tid & 15;   // fixed k column for softmax stage
  const int qr0 = tid >> 4;  // 0..7

  // stage K tiles for all 4 scales (persist across the whole q loop)
#pragma unroll
  for (int c = tid; c < 1024; c += 128) {
    const int p = c >> 8, r = (c >> 4) & 15, dd = (c & 15) << 2;
    const float4 v =
        *(const float4*)(Kp + ((size_t)(((bh << 2) + p) << 10) + k0 + r) * 64 + dd);
    KsS[p][r][dd] = v.x; KsS[p][r][dd + 1] = v.y;
    KsS[p][r][dd + 2] = v.z; KsS[p][r][dd + 3] = v.w;
  }
  float la0 = 0.f, la1 = 0.f, la2 = 0.f, la3 = 0.f;
  __syncthreads();

  for (int q0 = 0; q0 < 1024; q0 += 16) {
#pragma unroll
    for (int c = tid; c < 1024; c += 128) {
      const int p = c >> 8, r = (c >> 4) & 15, dd = (c & 15) << 2;
      const float4 v =
          *(const float4*)(Qp + ((size_t)(((bh << 2) + p) << 10) + q0 + r) * 64 + dd);
      QsS[p][r][dd] = v.x; QsS[p][r][dd + 1] = v.y;
      QsS[p][r][dd + 2] = v.z; QsS[p][r][dd + 3] = v.w;
    }
    // prefetch next q-tile (global_prefetch_b8); 64 rows x 2 half-row lines
    if (q0 + 16 < 1024) {
      const int pp = tid >> 5, rr = (tid >> 1) & 15, dp = (tid & 1) << 5;
      __builtin_prefetch(
          Qp + ((size_t)(((bh << 2) + pp) << 10) + q0 + 16 + rr) * 64 + dp, 0, 3);
    }
    __syncthreads();

    v8f acc = v8f_zero();
#pragma unroll
    for (int d0 = 0; d0 < 64; d0 += 4) {
      const v2f a = *(const v2f*)&QsS[wave][n][d0 + hoff];  // ds_load_b64
      const v2f b = *(const v2f*)&KsS[wave][n][d0 + hoff];  // ds_load_b64
      acc = wmma4(a, b, acc);
    }
#pragma unroll
    for (int i = 0; i < 8; ++i)
      Sx[wave][i + ((lane >> 4) << 3)][lane & 15] = acc[i];
    __syncthreads();

#pragma unroll
    for (int e = 0; e < 2; ++e) {
      const int qr = qr0 + (e << 3);
      const float s0 = Sx[0][qr][kc], s1 = Sx[1][qr][kc];
      const float s2 = Sx[2][qr][kc], s3 = Sx[3][qr][kc];
      const float mx = fmaxf(fmaxf(s0, s1), fmaxf(s2, s3));
      const float e0 = __expf((s0 - mx) * 0.125f);
      const float e1 = __expf((s1 - mx) * 0.125f);
      const float e2 = __expf((s2 - mx) * 0.125f);
      const float e3 = __expf((s3 - mx) * 0.125f);
      const float inv = 1.0f / (e0 + e1 + e2 + e3);
      la0 += e0 * inv; la1 += e1 * inv; la2 += e2 * inv; la3 += e3 * inv;
    }
    __syncthreads();
  }

  red[0][kc][qr0] = la0; red[1][kc][qr0] = la1;
  red[2][kc][qr0] = la2; red[3][kc][qr0] = la3;
  __syncthreads();
  if (tid < 64) {
    const int p = tid >> 4, c = tid & 15;
    float s = 0.f;
#pragma unroll
    for (int j = 0; j < 8; ++j) s += red[p][c][j];
    attn_out[(size_t)(((bh << 2) + p) << 10) + k0 + c] = s;
  }
}

__global__ void fill_zero_kernel(float* p, int nelem) {
  const int i = blockIdx.x * blockDim.x + threadIdx.x;
  if (i < nelem) p[i] = 0.f;
}

// ---------------------------------------------------------------------------
extern "C" void kernel_launch(void* const* d_in, const int* in_sizes, int n_in,
                              void* d_out, int out_size, void* d_ws, size_t ws_size,
                              hipStream_t stream) {
  (void)in_sizes; (void)n_in; (void)out_size;
  const float* Q = (const float*)d_in[0];
  const float* K = (const float*)d_in[1];
  const float* V = (const float*)d_in[2];
  const float* wq1 = (const float*)d_in[3];  const float* bq1 = (const float*)d_in[4];
  const float* wq3 = (const float*)d_in[5];  const float* bq3 = (const float*)d_in[6];
  const float* wq6 = (const float*)d_in[7];  const float* bq6 = (const float*)d_in[8];
  const float* wq9 = (const float*)d_in[9];  const float* bq9 = (const float*)d_in[10];
  const float* wk1 = (const float*)d_in[11]; const float* bk1 = (const float*)d_in[12];
  const float* wk3 = (const float*)d_in[13]; const float* bk3 = (const float*)d_in[14];
  const float* wk6 = (const float*)d_in[15]; const float* bk6 = (const float*)d_in[16];
  const float* wk9 = (const float*)d_in[17]; const float* bk9 = (const float*)d_in[18];

  float* ctx_out = (float*)d_out;
  float* attn_out = (float*)d_out + 2097152;

  // Workspace layout (float offsets); Qp reuses the dead Tc+Ts region.
  const size_t NEED_FLOATS = 23592960ull;  // ~90 MB
  if (ws_size < NEED_FLOATS * sizeof(float)) {
    // deterministic fallback: context is exact, attn_out zeroed
    context_kernel<<<32, 256, 0, stream>>>(V, ctx_out);
    fill_zero_kernel<<<(131072 + 255) / 256, 256, 0, stream>>>(attn_out, 131072);
    return;
  }
  float* ws = (float*)d_ws;
  float* CD  = ws;                //   262144
  float* SD  = ws + 262144;       //   262144
  float* CL  = ws + 524288;       //  1048576
  float* SLn = ws + 1572864;      //  1048576 (pre-negated S_L)
  float* Tc  = ws + 2621440;      //  8*524288
  float* Ts  = ws + 6815744;      //  8*524288
  float* F   = ws + 11010048;     //  8*524288  (Re FFT2, Q batches 0-3, K 4-7)
  float* Qp  = Tc;                //  8388608  (reuses Tc+Ts after pass 2)
  float* Kp  = ws + 15204352;     //  8388608

  // 1) DFT matrices
  build_dft_kernel<<<4096, 256, 0, stream>>>(CD, SD, CL, SLn);

  // 2) Tc/Ts = X * C_D / S_D  (M=1024, N=512, Kd=512; 4 batches per launch)
  dim3 g1(8, 16, 4);
  gemm2_kernel<<<g1, 128, 0, stream>>>(Q, 524288, CD, 0, nullptr, 0, nullptr, 0,
                                       Tc, 524288, 1024, 512, 512, 1);
  gemm2_kernel<<<g1, 128, 0, stream>>>(Q, 524288, SD, 0, nullptr, 0, nullptr, 0,
                                       Ts, 524288, 1024, 512, 512, 1);
  gemm2_kernel<<<g1, 128, 0, stream>>>(K, 524288, CD, 0, nullptr, 0, nullptr, 0,
                                       Tc + 4 * 524288, 524288, 1024, 512, 512, 1);
  gemm2_kernel<<<g1, 128, 0, stream>>>(K, 524288, SD, 0, nullptr, 0, nullptr, 0,
                                       Ts + 4 * 524288, 524288, 1024, 512, 512, 1);

  // 3) F = C_L*Tc + (-S_L)*Ts  (M=1024, N=512, Kd=1024; 8 batches)
  dim3 g2(8, 16, 8);
  gemm2_kernel<<<g2, 128, 0, stream>>>(CL, 0, Tc, 524288, SLn, 0, Ts, 524288,
                                       F, 524288, 1024, 512, 1024, 2);

  // 4) multi-scale windowed projections
  multiscale_kernel<<<32768, 256, 0, stream>>>(F, wq1, wq3, wq6, wq9,
                                               bq1, bq3, bq6, bq9, Qp);
  multiscale_kernel<<<32768, 256, 0, stream>>>(F + 4 * 524288, wk1, wk3, wk6, wk9,
                                               bk1, bk3, bk6, bk9, Kp);

  // 5) context = colsum(V) broadcast
  context_kernel<<<32, 256, 0, stream>>>(V, ctx_out);

  // 6) fused scores + softmax-over-p + sum-over-q
  attn_kernel<<<dim3(64, 32), 128, 0, stream>>>(Qp, Kp, attn_out);
}